// PrototypeClassifier_84894323573369
// MI455X (gfx1250) — compile-verified
//
#include <hip/hip_runtime.h>
#include <hip/hip_bf16.h>
#include <stdint.h>

// Problem constants (from reference): x[B,F], prototypes[C,F], out[B,C]
#define B_DIM 1024
#define C_DIM 512
#define F_DIM 1024

// Tiling
#define BM 64          // B rows per block
#define BN 32          // C cols per block
#define KF 32          // F chunk (floats) staged per iteration
#define NCHUNK (F_DIM / KF)
#define THREADS 128    // 4 waves (wave32)
#define LDS_STRIDE 44  // floats per LDS row: 176 B, 16B-aligned, bank-conflict-free padding

// fast sigmoid: 1 / (1 + exp2(-v * log2(e)))  -> v_exp_f32 + v_rcp_f32
__device__ __forceinline__ float fast_sigmoid(float v) {
    float e = __builtin_amdgcn_exp2f(-v * 1.44269504088896340736f);
    return __builtin_amdgcn_rcpf(1.0f + e);
}

// CDNA5 async global->LDS (ASYNCcnt-tracked). lds_off is the per-lane LDS byte
// address (low 32 bits of the generic shared-memory pointer; HW truncates the
// aperture bits), gaddr a per-lane 64-bit global address.
__device__ __forceinline__ void async_load_b128(uint32_t lds_off, const float* gaddr) {
    asm volatile("global_load_async_to_lds_b128 %0, %1, off"
                 :: "v"(lds_off), "v"(gaddr)
                 : "memory");
}

__device__ __forceinline__ void wait_asynccnt0() {
    asm volatile("s_wait_asynccnt 0x0" ::: "memory");
}

__global__ __launch_bounds__(THREADS)
void PrototypeClassifier_84894323573369_kernel(const float* __restrict__ x,
                                               const float* __restrict__ prot,
                                               float* __restrict__ out) {
    // Double-buffered tiles
    __shared__ float xs[2][BM * LDS_STRIDE];   // x tiles, 64 x 32 (padded)
    __shared__ float ps[2][BN * LDS_STRIDE];   // sigmoid(prototypes) tiles, 32 x 32 (padded)

    const int tid   = threadIdx.x;
    const int bRow0 = blockIdx.y * BM;      // 16 blocks in y
    const int cCol0 = blockIdx.x * BN;      // 16 blocks in x

    // Per-thread output micro-tile: 4 B-rows (contiguous) x 4 C-cols (stride 8)
    const int ty = tid >> 3;                // 0..15 -> rows r0..r0+3, r0 = 4*ty
    const int tx = tid & 7;                 // 0..7  -> cols tx, tx+8, tx+16, tx+24
    const int r0 = ty * 4;

    // Per-thread staging coordinates (reused every chunk)
    const int sRowX = tid >> 3;             // x tile: 8 float4 per row, 4 row-groups of 16
    const int sColX = (tid & 7) * 4;
    const int sRowP = tid >> 3;             // p tile: 8 float4 per row, 2 row-groups of 16
    const int sColP = (tid & 7) * 4;

    // LDS byte bases for async destination addresses
    const uint32_t xs_base0 = (uint32_t)(uintptr_t)(&xs[0][0]);
    const uint32_t xs_base1 = (uint32_t)(uintptr_t)(&xs[1][0]);

    float acc[4][4];
#pragma unroll
    for (int i = 0; i < 4; ++i)
#pragma unroll
        for (int j = 0; j < 4; ++j) acc[i][j] = 0.0f;

    // ---------------- staging helpers (inlined via lambdas) ----------------
    auto stage_x = [&](int buf, int k0) {
        const uint32_t base = buf ? xs_base1 : xs_base0;
#pragma unroll
        for (int t = 0; t < 4; ++t) {
            int row = sRowX + t * 16;                 // 0..63
            const float* g = x + (size_t)(bRow0 + row) * F_DIM + k0 + sColX;
            uint32_t loff  = base + (uint32_t)((row * LDS_STRIDE + sColX) * 4);
            async_load_b128(loff, g);
        }
    };
    auto stage_p = [&](int buf, int k0) {
#pragma unroll
        for (int t = 0; t < 2; ++t) {
            int row = sRowP + t * 16;                 // 0..31
            const float4 v = *(const float4*)(prot + (size_t)(cCol0 + row) * F_DIM + k0 + sColP);
            float4 s;
            s.x = fast_sigmoid(v.x);
            s.y = fast_sigmoid(v.y);
            s.z = fast_sigmoid(v.z);
            s.w = fast_sigmoid(v.w);
            *(float4*)(&ps[buf][row * LDS_STRIDE + sColP]) = s;   // ds_store_b128
        }
    };

    // ---------------- software pipeline: prologue ----------------
    stage_x(0, 0);
    stage_p(0, 0);

    for (int k = 0; k < NCHUNK; ++k) {
        const int cur = k & 1;

        wait_asynccnt0();      // this chunk's async x tile landed in LDS
        __syncthreads();       // + ds stores drained (implicit dscnt wait) & all waves ready

        // Prefetch next chunk into the other buffer while we compute this one.
        if (k + 1 < NCHUNK) {
            stage_x(cur ^ 1, (k + 1) * KF);
            stage_p(cur ^ 1, (k + 1) * KF);
        }

        // ---- compute: 8 steps of 4 f-values via float4 LDS loads ----
        const float* xbuf = &xs[cur][0];
        const float* pbuf = &ps[cur][0];
#pragma unroll
        for (int kk = 0; kk < KF; kk += 4) {
            float4 xv[4], pv[4];
#pragma unroll
            for (int i = 0; i < 4; ++i)
                xv[i] = *(const float4*)(&xbuf[(r0 + i) * LDS_STRIDE + kk]);
#pragma unroll
            for (int j = 0; j < 4; ++j)
                pv[j] = *(const float4*)(&pbuf[(tx + 8 * j) * LDS_STRIDE + kk]);
#pragma unroll
            for (int i = 0; i < 4; ++i)
#pragma unroll
                for (int j = 0; j < 4; ++j) {
                    acc[i][j] += fabsf(xv[i].x - pv[j].x);
                    acc[i][j] += fabsf(xv[i].y - pv[j].y);
                    acc[i][j] += fabsf(xv[i].z - pv[j].z);
                    acc[i][j] += fabsf(xv[i].w - pv[j].w);
                }
        }
        // No trailing barrier needed: next iteration's wait+barrier orders the
        // overwrite of buf cur (it is only re-staged at iteration k+1, after
        // every wave has finished computing from it at iteration k).
    }

    // ---- write out[b, c], c strided by 8 within the 32-wide tile ----
#pragma unroll
    for (int i = 0; i < 4; ++i) {
        const size_t rowOff = (size_t)(bRow0 + r0 + i) * C_DIM + cCol0;
#pragma unroll
        for (int j = 0; j < 4; ++j) {
            out[rowOff + tx + 8 * j] = acc[i][j];
        }
    }
}

extern "C" void kernel_launch(void* const* d_in, const int* in_sizes, int n_in,
                              void* d_out, int out_size, void* d_ws, size_t ws_size,
                              hipStream_t stream) {
    (void)in_sizes; (void)n_in; (void)d_ws; (void)ws_size; (void)out_size;
    const float* x    = (const float*)d_in[0];
    const float* prot = (const float*)d_in[1];
    float* out        = (float*)d_out;

    dim3 grid(C_DIM / BN, B_DIM / BM);   // (16, 16) = 256 blocks
    dim3 block(THREADS);                 // 128 threads = 4 wave32
    PrototypeClassifier_84894323573369_kernel<<<grid, block, 0, stream>>>(x, prot, out);
}